// GTDetector_3693671875315
// MI455X (gfx1250) — compile-verified
//
#include <hip/hip_runtime.h>

typedef __attribute__((ext_vector_type(2))) float v2f;
typedef __attribute__((ext_vector_type(4))) float v4f;
typedef __attribute__((ext_vector_type(8))) float v8f;

#define IN_CH   134
#define MOUT    16
#define ROWS_PER_BLOCK 128   // 8 waves * 16 rows
#define KFULL   132          // kb = 0..128 step 4: all K-pairs in-bounds

// ---------------------------------------------------------------------------
// Kernel 1: feats = relu(voxel_feats @ W)   [num_voxels x 134] x [134 x 16]
// One wave per 16-voxel tile, V_WMMA_F32_16X16X4_F32 over K.
// A staged through LDS (coalesced NT 16B streaming), W resident in LDS.
// Tail K-step (kb=132) peeled so the hot loop is branch-free and unrollable.
// ---------------------------------------------------------------------------
__global__ void __launch_bounds__(256) gemm_relu_wmma(
    const float* __restrict__ vf,   // [num_voxels, 134]
    const float* __restrict__ W,    // [134, 16]
    float* __restrict__ feats,      // [num_voxels, 16]
    int num_voxels)
{
    __shared__ float Wl[(KFULL + 4) * MOUT];     // K-padded (rows 134,135 zero)
    __shared__ float As[ROWS_PER_BLOCK * IN_CH]; // 67 KB, A tile (row stride 134)

    const int tid = threadIdx.x;

    // --- load W into LDS, zero the K-pad rows ---
    for (int i = tid; i < (KFULL + 4) * MOUT; i += 256)
        Wl[i] = (i < IN_CH * MOUT) ? W[i] : 0.0f;

    // --- stream A tile into LDS: coalesced non-temporal 16B loads ---
    const int row_base = blockIdx.x * ROWS_PER_BLOCK;
    const int rows = min(ROWS_PER_BLOCK, num_voxels - row_base);
    if (rows > 0) {
        const float* gsrc = vf + (long)row_base * IN_CH;
        const int n_elems = rows * IN_CH;        // rows even -> divisible by 4
        const int n4 = n_elems >> 2;
        const v4f* gsrc4 = (const v4f*)gsrc;
        v4f* lds4 = (v4f*)As;
        for (int i = tid; i < n4; i += 256)
            lds4[i] = __builtin_nontemporal_load(&gsrc4[i]);
        for (int i = (n4 << 2) + tid; i < n_elems; i += 256)  // safety tail
            As[i] = __builtin_nontemporal_load(&gsrc[i]);
    }
    __syncthreads();

    const int wave   = tid >> 5;                 // 0..7
    const int lane   = tid & 31;
    const int lrow   = lane & 15;                // M row of A / N col of B,C,D
    const int h      = lane >> 4;                // K-pair selector
    const int tile_row = wave * 16;
    if (tile_row >= rows) return;                // partial last block

    const float* arow = As + (tile_row + lrow) * IN_CH;

    v8f acc = {};
    // Branch-free main loop: every A pair (k0, k0+1) is in-bounds.
    #pragma unroll
    for (int kb = 0; kb < KFULL; kb += 4) {
        const int k0 = kb + 2 * h;
        v2f a; a.x = arow[k0]; a.y = arow[k0 + 1];
        v2f b;
        b.x = Wl[(k0 + 0) * MOUT + lrow];
        b.y = Wl[(k0 + 1) * MOUT + lrow];
        acc = __builtin_amdgcn_wmma_f32_16x16x4_f32(
            false, a, false, b, (short)0, acc, false, false);
    }
    // Peeled tail: kb = 132 -> k0 in {132, 134}; h==1 lanes contribute zero.
    {
        const int k0 = KFULL + 2 * h;            // 132 or 134
        v2f a;
        if (h == 0) { a.x = arow[k0]; a.y = arow[k0 + 1]; }
        else        { a.x = 0.0f;     a.y = 0.0f; }
        v2f b;                                    // Wl rows 134,135 are zero
        b.x = Wl[(k0 + 0) * MOUT + lrow];
        b.y = Wl[(k0 + 1) * MOUT + lrow];
        acc = __builtin_amdgcn_wmma_f32_16x16x4_f32(
            false, a, false, b, (short)0, acc, false, false);
    }

    // C/D layout: VGPR r -> row (r + 8*h), lane&15 -> col. Apply ReLU, store.
    const long out_row0 = (long)(row_base + tile_row);
    #pragma unroll
    for (int r = 0; r < 8; ++r) {
        float v = acc[r];
        v = v > 0.0f ? v : 0.0f;
        feats[(out_row0 + r + 8 * h) * MOUT + lrow] = v;
    }
}

// ---------------------------------------------------------------------------
// Kernel 2: per-proposal segment mean + batchId + objectness.
// One block per proposal. Thread layout: 16 point-groups x 16 channels, so
// each point's 16-float feature row is one coalesced 64B read (L2-resident).
// ---------------------------------------------------------------------------
__global__ void __launch_bounds__(256) proposal_reduce(
    const float* __restrict__ feats,     // [num_voxels, 16]
    const int*   __restrict__ p2v_map,   // [N_POINTS]
    const int*   __restrict__ gt_idx,    // [T, 2] flat
    const int*   __restrict__ gt_off,    // [P+1]
    const int*   __restrict__ locs,      // [N_POINTS, 4] flat
    float* __restrict__ out,             // [P*16 | P | P]
    int P, int T)
{
    const int p = blockIdx.x;
    if (p >= P) return;

    const int beg   = gt_off[p];
    const int end   = gt_off[p + 1];
    const int count = end - beg;

    const int tid = threadIdx.x;
    const int ch  = tid & 15;    // channel
    const int grp = tid >> 4;    // point-group 0..15

    float acc = 0.0f;
    for (int t = beg + grp; t < end; t += 16) {
        const int pi = gt_idx[2 * t + 1];
        const int v  = p2v_map[pi];
        acc += feats[(long)v * MOUT + ch];
    }

    __shared__ float red[256];
    red[tid] = acc;
    __syncthreads();
    #pragma unroll
    for (int s = 8; s > 0; s >>= 1) {
        if (grp < s) red[grp * 16 + ch] += red[(grp + s) * 16 + ch];
        __syncthreads();
    }

    if (grp == 0) {
        const float denom = (count > 0) ? (float)count : 1.0f;
        out[p * MOUT + ch] = red[ch] / denom;
    }
    if (tid == 0) {
        int t0 = beg < T ? beg : (T - 1);        // clamp like jax gather
        if (t0 < 0) t0 = 0;
        const int pi0 = gt_idx[2 * t0 + 1];
        out[P * MOUT + p]     = (float)locs[4 * pi0 + 0];  // batchId
        out[P * MOUT + P + p] = 1.0f;                      // objectness
    }
}

// ---------------------------------------------------------------------------
extern "C" void kernel_launch(void* const* d_in, const int* in_sizes, int n_in,
                              void* d_out, int out_size, void* d_ws, size_t ws_size,
                              hipStream_t stream)
{
    const float* vf     = (const float*)d_in[0];   // voxel_feats
    const float* W      = (const float*)d_in[1];   // W
    const int*   p2v    = (const int*)  d_in[2];   // p2v_map
    const int*   gt_idx = (const int*)  d_in[3];   // gt_proposals_idx (T,2)
    const int*   gt_off = (const int*)  d_in[4];   // gt_proposals_offset (P+1)
    const int*   locs   = (const int*)  d_in[5];   // locs_scaled (N,4)
    float* out   = (float*)d_out;
    float* feats = (float*)d_ws;                   // [num_voxels,16] f32 = 64 MB

    const int num_voxels = in_sizes[0] / IN_CH;
    const int T          = in_sizes[3] / 2;
    const int P          = in_sizes[4] - 1;

    const int blocks1 = (num_voxels + ROWS_PER_BLOCK - 1) / ROWS_PER_BLOCK;
    gemm_relu_wmma<<<blocks1, 256, 0, stream>>>(vf, W, feats, num_voxels);
    proposal_reduce<<<P, 256, 0, stream>>>(feats, p2v, gt_idx, gt_off, locs,
                                           out, P, T);
}